// RobustTripletLoss_40114994544728
// MI455X (gfx1250) — compile-verified
//
#include <hip/hip_runtime.h>

typedef float v2f __attribute__((ext_vector_type(2)));
typedef float v4f __attribute__((ext_vector_type(4)));
typedef float v8f __attribute__((ext_vector_type(8)));

#define DIMS 512
#define HALF_K (DIMS / 2)   // 256: lo-half lanes sweep [0,256), hi-half [256,512)
#define BIGF 1.0e4f

__global__ void init_acc_kernel(float* acc) {
    acc[0] = 0.0f;
    acc[1] = 0.0f;
}

// One block per row: L2-normalize 512 fp32 values (eps=1e-12 clamp on the norm).
__global__ __launch_bounds__(256) void normalize_rows_kernel(const float* __restrict__ E,
                                                             float* __restrict__ En) {
    const int row = blockIdx.x;
    const int tid = threadIdx.x;
    const float2* src = (const float2*)(E + (size_t)row * DIMS);
    float2 v = src[tid];                       // 256 threads x float2 = 512 floats
    float ss = v.x * v.x + v.y * v.y;
#pragma unroll
    for (int off = 16; off >= 1; off >>= 1) ss += __shfl_xor(ss, off, 32);
    __shared__ float wsum[8];
    if ((tid & 31) == 0) wsum[tid >> 5] = ss;
    __syncthreads();
    float tot = 0.0f;
#pragma unroll
    for (int w = 0; w < 8; ++w) tot += wsum[w];
    const float scale = 1.0f / fmaxf(sqrtf(tot), 1e-12f);
    float2 o;
    o.x = v.x * scale;
    o.y = v.y * scale;
    ((float2*)(En + (size_t)row * DIMS))[tid] = o;
}

// One workgroup (8 waves) per 16 output rows. Fused: WMMA fp32 GEMM tile ->
// cosine distance -> masked row max/min (batch-hard mining) -> block atomics.
//
// K-slice trick: WMMA step s consumes K = {2s, 2s+1} on lanes 0-15 and
// K = {256+2s, 256+2s+1} on lanes 16-31 (any partition of K is valid as long
// as A and B fragments agree). This makes every lane's A/B stream contiguous,
// so one b128 load supplies TWO WMMA steps (2 WMMAs per global load per
// accumulator). Four accumulators per wave give four independent load streams.
__global__ __launch_bounds__(256) void hard_mine_tiles_kernel(const float* __restrict__ En,
                                                              const int* __restrict__ labels,
                                                              int B,
                                                              float* __restrict__ acc) {
    __shared__ __align__(16) float As[16 * DIMS];   // 32 KB A tile (16 rows x 512)
    __shared__ float pMax[8][16];
    __shared__ float pMin[8][16];
    __shared__ float red[32];

    const int tid     = threadIdx.x;
    const int wave    = tid >> 5;
    const int lane    = tid & 31;
    const int hi      = lane >> 4;   // half-wave: selects K half / M group
    const int l16     = lane & 15;
    const int rowBase = blockIdx.x * 16;

    // Stage A tile into LDS (float4 cooperative copy: 2048 float4 / 256 threads).
    {
        const float4* g = (const float4*)(En + (size_t)rowBase * DIMS);
        float4* s = (float4*)As;
#pragma unroll
        for (int i = 0; i < (16 * DIMS / 4) / 256; ++i)
            s[tid + i * 256] = g[tid + i * 256];
    }

    // Labels of the 8 M-rows this lane's C-fragment elements belong to.
    int rl[8];
#pragma unroll
    for (int v = 0; v < 8; ++v) rl[v] = labels[rowBase + 8 * hi + v];

    float maxP[8], minN[8];
#pragma unroll
    for (int v = 0; v < 8; ++v) { maxP[v] = -1.0f; minN[v] = BIGF; }

    __syncthreads();

    const int nTiles = B >> 4;                       // 256 column tiles of 16
    for (int jt = wave * 4; jt < nTiles; jt += 32) { // each wave: 4 N-tiles / iter
        const int cb0 = (jt + 0) * 16;
        const int cb1 = (jt + 1) * 16;
        const int cb2 = (jt + 2) * 16;
        const int cb3 = (jt + 3) * 16;
        const int cl0 = labels[cb0 + l16];
        const int cl1 = labels[cb1 + l16];
        const int cl2 = labels[cb2 + l16];
        const int cl3 = labels[cb3 + l16];
        const float* ap  = As + l16 * DIMS + hi * HALF_K;
        const float* b0p = En + (size_t)(cb0 + l16) * DIMS + hi * HALF_K;
        const float* b1p = En + (size_t)(cb1 + l16) * DIMS + hi * HALF_K;
        const float* b2p = En + (size_t)(cb2 + l16) * DIMS + hi * HALF_K;
        const float* b3p = En + (size_t)(cb3 + l16) * DIMS + hi * HALF_K;

        v8f c0 = {0.f, 0.f, 0.f, 0.f, 0.f, 0.f, 0.f, 0.f};
        v8f c1 = c0, c2 = c0, c3 = c0;

        // 128 WMMA steps per accumulator; each b128 load feeds 2 steps.
#pragma unroll 2
        for (int t = 0; t < HALF_K; t += 4) {
            v4f a  = *(const v4f*)(ap  + t);
            v4f b0 = *(const v4f*)(b0p + t);
            v4f b1 = *(const v4f*)(b1p + t);
            v4f b2 = *(const v4f*)(b2p + t);
            v4f b3 = *(const v4f*)(b3p + t);

            v2f aL = __builtin_shufflevector(a, a, 0, 1);
            v2f aH = __builtin_shufflevector(a, a, 2, 3);

            c0 = __builtin_amdgcn_wmma_f32_16x16x4_f32(false, aL, false,
                     __builtin_shufflevector(b0, b0, 0, 1), (short)0, c0, false, false);
            c1 = __builtin_amdgcn_wmma_f32_16x16x4_f32(false, aL, false,
                     __builtin_shufflevector(b1, b1, 0, 1), (short)0, c1, false, false);
            c2 = __builtin_amdgcn_wmma_f32_16x16x4_f32(false, aL, false,
                     __builtin_shufflevector(b2, b2, 0, 1), (short)0, c2, false, false);
            c3 = __builtin_amdgcn_wmma_f32_16x16x4_f32(false, aL, false,
                     __builtin_shufflevector(b3, b3, 0, 1), (short)0, c3, false, false);

            c0 = __builtin_amdgcn_wmma_f32_16x16x4_f32(false, aH, false,
                     __builtin_shufflevector(b0, b0, 2, 3), (short)0, c0, false, false);
            c1 = __builtin_amdgcn_wmma_f32_16x16x4_f32(false, aH, false,
                     __builtin_shufflevector(b1, b1, 2, 3), (short)0, c1, false, false);
            c2 = __builtin_amdgcn_wmma_f32_16x16x4_f32(false, aH, false,
                     __builtin_shufflevector(b2, b2, 2, 3), (short)0, c2, false, false);
            c3 = __builtin_amdgcn_wmma_f32_16x16x4_f32(false, aH, false,
                     __builtin_shufflevector(b3, b3, 2, 3), (short)0, c3, false, false);
        }

        // C/D layout: lane holds N=l16, element v is M = v + 8*hi.
        const int gc0 = cb0 + l16;
        const int gc1 = cb1 + l16;
        const int gc2 = cb2 + l16;
        const int gc3 = cb3 + l16;
#pragma unroll
        for (int v = 0; v < 8; ++v) {
            const int gr = rowBase + 8 * hi + v;

            float d0 = fmaxf(1.0f - c0[v], 0.0f);
            bool same0 = (rl[v] == cl0);
            if (same0 && gr != gc0) maxP[v] = fmaxf(maxP[v], d0);
            if (!same0)             minN[v] = fminf(minN[v], d0);

            float d1 = fmaxf(1.0f - c1[v], 0.0f);
            bool same1 = (rl[v] == cl1);
            if (same1 && gr != gc1) maxP[v] = fmaxf(maxP[v], d1);
            if (!same1)             minN[v] = fminf(minN[v], d1);

            float d2 = fmaxf(1.0f - c2[v], 0.0f);
            bool same2 = (rl[v] == cl2);
            if (same2 && gr != gc2) maxP[v] = fmaxf(maxP[v], d2);
            if (!same2)             minN[v] = fminf(minN[v], d2);

            float d3 = fmaxf(1.0f - c3[v], 0.0f);
            bool same3 = (rl[v] == cl3);
            if (same3 && gr != gc3) maxP[v] = fmaxf(maxP[v], d3);
            if (!same3)             minN[v] = fminf(minN[v], d3);
        }
    }

    // Reduce over the 16 lanes that share each M row (butterfly within half-wave).
#pragma unroll
    for (int off = 8; off >= 1; off >>= 1) {
#pragma unroll
        for (int v = 0; v < 8; ++v) {
            maxP[v] = fmaxf(maxP[v], __shfl_xor(maxP[v], off, 16));
            minN[v] = fminf(minN[v], __shfl_xor(minN[v], off, 16));
        }
    }
    if (l16 == 0) {
#pragma unroll
        for (int v = 0; v < 8; ++v) {
            pMax[wave][8 * hi + v] = maxP[v];
            pMin[wave][8 * hi + v] = minN[v];
        }
    }
    __syncthreads();

    // Combine the 8 waves' partials (each wave covered a disjoint set of columns).
    if (tid < 16) {
        float mp = -1.0f, mn = BIGF;
#pragma unroll
        for (int w = 0; w < 8; ++w) {
            mp = fmaxf(mp, pMax[w][tid]);
            mn = fminf(mn, pMin[w][tid]);
        }
        const bool valid = (mp >= 0.0f) && (mn < BIGF);
        const float per  = fmaxf(fmaxf(mp, 0.0f) - mn + 1.0f, 0.0f);
        red[tid]      = valid ? per  : 0.0f;
        red[16 + tid] = valid ? 1.0f : 0.0f;
    }
    __syncthreads();
    if (tid == 0) {
        float s = 0.0f, c = 0.0f;
#pragma unroll
        for (int i = 0; i < 16; ++i) { s += red[i]; c += red[16 + i]; }
        atomicAdd(&acc[0], s);
        atomicAdd(&acc[1], c);
    }
}

__global__ void finalize_kernel(const float* acc, float* out) {
    out[0] = acc[0] / fmaxf(acc[1], 1.0f);
}

extern "C" void kernel_launch(void* const* d_in, const int* in_sizes, int n_in,
                              void* d_out, int out_size, void* d_ws, size_t ws_size,
                              hipStream_t stream) {
    (void)n_in; (void)out_size; (void)ws_size;
    const float* E      = (const float*)d_in[0];   // [B, 512] fp32
    const int*   labels = (const int*)d_in[1];     // [B] int
    const int B = in_sizes[1];                     // 4096

    float* W   = (float*)d_ws;
    float* En  = W;                                // [B, 512] normalized embeddings
    float* acc = W + (size_t)B * DIMS;             // {loss_sum, num_valid}

    init_acc_kernel<<<1, 1, 0, stream>>>(acc);
    normalize_rows_kernel<<<B, 256, 0, stream>>>(E, En);
    hard_mine_tiles_kernel<<<B / 16, 256, 0, stream>>>(En, labels, B, acc);
    finalize_kernel<<<1, 1, 0, stream>>>(acc, (float*)d_out);
}